// PoissonMatchLoss_56418690400577
// MI455X (gfx1250) — compile-verified
//
#include <hip/hip_runtime.h>
#include <math.h>
#include <stdint.h>

// Problem geometry (from reference): D = 180 detector bins.
#define D_DIM 180
#define NQ    (D_DIM / 4)   // 45 float4s per row
#define LSTR  188           // padded LDS row stride in floats:
                            //  - 188*4 % 16 == 0 -> float4/b128-aligned rows
                            //  - 188 % 64 == 60 = 4*15; 15 coprime 16 -> 16 distinct
                            //    bank groups for the 16-lane fragment loads
#define ROWS  32            // i-rows per block (2 x 16 subtiles, register-blocked)
#define COLS  128           // j-cols per staged chunk (4 waves x 2 x 16 subtiles)
#define NK    (D_DIM / 4)   // 45 K=4 WMMA steps

#define INF_F 3.402823466e+38f
#define LOG_2PI 1.8378770664093453f

typedef float v2f __attribute__((ext_vector_type(2)));
typedef float v4f __attribute__((ext_vector_type(4)));
typedef float v8f __attribute__((ext_vector_type(8)));

// ---------------------------------------------------------------------------
// Kernel 1: per-row sums.  rows [0,N) -> s_in;  rows [N,N+M) -> Stirling sums.
// One wave32 per row, shuffle reduction.
// ---------------------------------------------------------------------------
__global__ __launch_bounds__(256)
void rowstats_kernel(const float* __restrict__ inp,
                     const float* __restrict__ tgt,
                     float* __restrict__ s_in,
                     float* __restrict__ s_st,
                     int N, int M) {
  const int wave = threadIdx.x >> 5;
  const int lane = threadIdx.x & 31;
  const int row  = blockIdx.x * 8 + wave;

  if (row < N) {
    const float* p = inp + (size_t)row * D_DIM;
    float s = 0.f;
    for (int d = lane; d < D_DIM; d += 32) s += p[d];
    #pragma unroll
    for (int m = 16; m >= 1; m >>= 1) s += __shfl_xor(s, m, 32);
    if (lane == 0) s_in[row] = s;
  } else if (row < N + M) {
    const int j = row - N;
    const float* p = tgt + (size_t)j * D_DIM;
    float s = 0.f;
    for (int d = lane; d < D_DIM; d += 32) {
      float t = p[d];
      if (t > 1.f) {
        float lt = logf(t);
        s += t * lt - t + 0.5f * (LOG_2PI + lt);   // t*log t - t + 0.5*log(2*pi*t)
      }
    }
    #pragma unroll
    for (int m = 16; m >= 1; m >>= 1) s += __shfl_xor(s, m, 32);
    if (lane == 0) s_st[j] = s;
  }
}

// ---------------------------------------------------------------------------
// Kernel 2: fused  cross = log(inp+eps) @ tgt^T  +  rowwise min/argmin of
//           L[i,j] = s_in[i] - cross[i,j] + s_st[j]
// f32 WMMA (16x16x4), 2x2 register blocking per wave (1 ds_load_b64 per WMMA).
// B chunks staged via GLOBAL_LOAD_ASYNC_TO_LDS_B128 (ASYNCcnt path): L2->LDS
// direct, no VGPR transit; ~118 KB LDS/block -> 2 blocks per 320 KB WGP, so
// one block's compute overlaps the other's staging.
// ---------------------------------------------------------------------------
__global__ __launch_bounds__(128)
void poisson_pair_min_kernel(const float* __restrict__ inp,
                             const float* __restrict__ tgt,
                             const float* __restrict__ s_in,
                             const float* __restrict__ s_st,
                             float* __restrict__ loss_v,
                             int*   __restrict__ match,
                             int N, int M) {
  __shared__ float Alds[ROWS * LSTR];   // 23.5 KB
  __shared__ float Blds[COLS * LSTR];   // 94 KB
  __shared__ float mergeV[ROWS][4];
  __shared__ int   mergeJ[ROWS][4];

  const int tid   = threadIdx.x;
  const int ibase = blockIdx.x * ROWS;

  // ---- Stage A tile: log(inp + eps), float4 loads/stores (needs VALU) ----
  for (int fid = tid; fid < ROWS * NQ; fid += 128) {
    const int r  = fid / NQ;
    const int q  = fid % NQ;
    const int gi = ibase + r;
    v4f v = {0.f, 0.f, 0.f, 0.f};
    if (gi < N) {
      v4f x = *(const v4f*)(inp + (size_t)gi * D_DIM + 4 * q);
      v.x = logf(x.x + 1e-8f);
      v.y = logf(x.y + 1e-8f);
      v.z = logf(x.z + 1e-8f);
      v.w = logf(x.w + 1e-8f);
    }
    *(v4f*)(Alds + r * LSTR + 4 * q) = v;
  }

  const int wave = tid >> 5;        // 0..3  -> 32-col slice within the chunk
  const int lane = tid & 31;
  const int half = lane >> 4;       // ISA half-wave: K/M offset selector
  const int l16  = lane & 15;

  // Fragment base pointers (ISA 7.12.2: 32-bit A 16x4, mirrored B 4x16):
  //   lanes 0-15 hold K = k0+0,k0+1 ; lanes 16-31 hold K = k0+2,k0+3
  const float* AfragR0 = Alds + (l16     ) * LSTR + half * 2;  // rows  0..15
  const float* AfragR1 = Alds + (l16 + 16) * LSTR + half * 2;  // rows 16..31
  const float* BfragC0 = Blds + (wave * 32 + l16     ) * LSTR + half * 2;
  const float* BfragC1 = Blds + (wave * 32 + l16 + 16) * LSTR + half * 2;

  float minv0[8], minv1[8];
  int   minj0[8], minj1[8];
  #pragma unroll
  for (int r = 0; r < 8; ++r) {
    minv0[r] = INF_F; minj0[r] = 0;
    minv1[r] = INF_F; minj1[r] = 0;
  }

  for (int jbase = 0; jbase < M; jbase += COLS) {
    __syncthreads();   // previous chunk's fragment reads done before overwrite

    // ---- Stage B chunk via async global->LDS DMA (no VGPR transit) ----
    {
      const float* chunk = tgt + (size_t)jbase * D_DIM;   // uniform -> SGPR pair
      for (int fid = tid; fid < COLS * NQ; fid += 128) {
        const int c  = fid / NQ;
        const int q  = fid % NQ;
        const int gj = jbase + c;
        if (gj < M) {
          // LDS aperture keeps the byte offset in addr[31:0]
          uint32_t ldsoff = (uint32_t)(uintptr_t)(Blds + c * LSTR + 4 * q);
          uint32_t goff   = (uint32_t)((c * D_DIM + 4 * q) * 4);
          asm volatile(
            "global_load_async_to_lds_b128 %0, %1, %2 offset:0"
            :
            : "v"(ldsoff), "v"(goff), "s"(chunk)
            : "memory");
        } else {
          v4f z = {0.f, 0.f, 0.f, 0.f};                   // OOB cols -> zeros
          *(v4f*)(Blds + c * LSTR + 4 * q) = z;
        }
      }
      asm volatile("s_wait_asynccnt 0x0" ::: "memory");
    }
    // hint next chunk into cache while we compute this one
    if (jbase + COLS + tid < M)
      __builtin_prefetch(tgt + (size_t)(jbase + COLS + tid) * D_DIM, 0, 0);
    __syncthreads();

    const int   jc0  = jbase + wave * 32 + l16;          // cols of C*0 tiles
    const int   jc1  = jc0 + 16;                         // cols of C*1 tiles
    const float sst0 = (jc0 < M) ? s_st[jc0] : INF_F;    // OOB never wins min
    const float sst1 = (jc1 < M) ? s_st[jc1] : INF_F;

    v8f C00 = {}, C01 = {}, C10 = {}, C11 = {};
    #pragma unroll
    for (int kk = 0; kk < NK; ++kk) {
      v2f a0 = *(const v2f*)(AfragR0 + kk * 4);
      v2f a1 = *(const v2f*)(AfragR1 + kk * 4);
      v2f b0 = *(const v2f*)(BfragC0 + kk * 4);
      v2f b1 = *(const v2f*)(BfragC1 + kk * 4);
      C00 = __builtin_amdgcn_wmma_f32_16x16x4_f32(false, a0, false, b0,
                                                  (short)0, C00, false, false);
      C01 = __builtin_amdgcn_wmma_f32_16x16x4_f32(false, a0, false, b1,
                                                  (short)0, C01, false, false);
      C10 = __builtin_amdgcn_wmma_f32_16x16x4_f32(false, a1, false, b0,
                                                  (short)0, C10, false, false);
      C11 = __builtin_amdgcn_wmma_f32_16x16x4_f32(false, a1, false, b1,
                                                  (short)0, C11, false, false);
    }

    // running min/argmin of (s_st[j] - cross[i,j]); lower-j tile first so
    // strict '<' preserves jnp.argmin first-minimum semantics
    #pragma unroll
    for (int r = 0; r < 8; ++r) {
      float v;
      v = sst0 - C00[r]; if (v < minv0[r]) { minv0[r] = v; minj0[r] = jc0; }
      v = sst1 - C01[r]; if (v < minv0[r]) { minv0[r] = v; minj0[r] = jc1; }
      v = sst0 - C10[r]; if (v < minv1[r]) { minv1[r] = v; minj1[r] = jc0; }
      v = sst1 - C11[r]; if (v < minv1[r]) { minv1[r] = v; minj1[r] = jc1; }
    }
  }

  // ---- butterfly reduce across the 16 lanes of each half (ties -> lower j) ----
  #pragma unroll
  for (int r = 0; r < 8; ++r) {
    #pragma unroll
    for (int m = 1; m <= 8; m <<= 1) {
      float ov; int oj;
      ov = __shfl_xor(minv0[r], m, 32);
      oj = __shfl_xor(minj0[r], m, 32);
      if (ov < minv0[r] || (ov == minv0[r] && oj < minj0[r])) {
        minv0[r] = ov; minj0[r] = oj;
      }
      ov = __shfl_xor(minv1[r], m, 32);
      oj = __shfl_xor(minj1[r], m, 32);
      if (ov < minv1[r] || (ov == minv1[r] && oj < minj1[r])) {
        minv1[r] = ov; minj1[r] = oj;
      }
    }
  }
  if (l16 == 0) {      // lanes 0 (rows r) and 16 (rows r+8) publish per wave
    #pragma unroll
    for (int r = 0; r < 8; ++r) {
      const int row0 = half * 8 + r;        // subtile 0: rows  0..15
      const int row1 = 16 + half * 8 + r;   // subtile 1: rows 16..31
      mergeV[row0][wave] = minv0[r];  mergeJ[row0][wave] = minj0[r];
      mergeV[row1][wave] = minv1[r];  mergeJ[row1][wave] = minj1[r];
    }
  }
  __syncthreads();

  // ---- merge the 4 wave slices ((value, j) lexicographic), finalize row ----
  if (tid < ROWS) {
    float v0 = mergeV[tid][0]; int j0 = mergeJ[tid][0];
    #pragma unroll
    for (int g = 1; g < 4; ++g) {
      float v1 = mergeV[tid][g]; int j1 = mergeJ[tid][g];
      if (v1 < v0 || (v1 == v0 && j1 < j0)) { v0 = v1; j0 = j1; }
    }
    const int gi = ibase + tid;
    if (gi < N) {
      loss_v[gi] = s_in[gi] + v0;   // L = s_in - cross + s_st
      match[gi]  = j0;
    }
  }
}

// ---------------------------------------------------------------------------
// Kernel 3: loss = mean(loss_v)
// ---------------------------------------------------------------------------
__global__ __launch_bounds__(256)
void mean_kernel(const float* __restrict__ loss_v, float* __restrict__ out, int N) {
  __shared__ float red[256];
  float s = 0.f;
  for (int i = threadIdx.x; i < N; i += 256) s += loss_v[i];
  red[threadIdx.x] = s;
  __syncthreads();
  for (int o = 128; o > 0; o >>= 1) {
    if (threadIdx.x < o) red[threadIdx.x] += red[threadIdx.x + o];
    __syncthreads();
  }
  if (threadIdx.x == 0) out[0] = red[0] / (float)N;
}

// ---------------------------------------------------------------------------
extern "C" void kernel_launch(void* const* d_in, const int* in_sizes, int n_in,
                              void* d_out, int out_size, void* d_ws, size_t ws_size,
                              hipStream_t stream) {
  const float* inp = (const float*)d_in[0];
  const float* tgt = (const float*)d_in[1];
  const int N = in_sizes[0] / D_DIM;
  const int M = in_sizes[1] / D_DIM;

  float* ws     = (float*)d_ws;
  float* s_in   = ws;             // [N]
  float* s_st   = ws + N;         // [M]
  float* loss_v = ws + N + M;     // [N]

  float* outf  = (float*)d_out;        // scalar loss at d_out[0]
  int*   match = (int*)d_out + 1;      // argmin indices (int32 bits)

  const int rsBlocks = (N + M + 7) / 8;
  rowstats_kernel<<<rsBlocks, 256, 0, stream>>>(inp, tgt, s_in, s_st, N, M);

  const int pBlocks = (N + ROWS - 1) / ROWS;
  poisson_pair_min_kernel<<<pBlocks, 128, 0, stream>>>(inp, tgt, s_in, s_st,
                                                       loss_v, match, N, M);

  mean_kernel<<<1, 256, 0, stream>>>(loss_v, outf, N);
}